// VirtualNode_21028159881544
// MI455X (gfx1250) — compile-verified
//
#include <hip/hip_runtime.h>

typedef __attribute__((ext_vector_type(2))) float v2f;
typedef __attribute__((ext_vector_type(4))) float f4;
typedef __attribute__((ext_vector_type(8))) float v8f;

#define DIM 128
#define NPW 256   // nodes per wave in segment-sum

// ---------------- kernel 0: zero the pool accumulator ----------------
__global__ void vn_zero_pool(float* __restrict__ pool, int n) {
    int i = blockIdx.x * blockDim.x + threadIdx.x;
    if (i < n) pool[i] = 0.0f;
}

// ---------------- kernel 1: ragged segment sum (sorted ids) ----------
// One wave handles NPW contiguous node rows; lane l accumulates cols
// [4l, 4l+4) in registers and flushes with f32 atomics on id change.
__global__ void vn_segsum(const float* __restrict__ h,
                          const int* __restrict__ seg,
                          float* __restrict__ pool, int n) {
    int wavesPerBlock = blockDim.x >> 5;
    int wave = blockIdx.x * wavesPerBlock + (threadIdx.x >> 5);
    int lane = threadIdx.x & 31;
    long start = (long)wave * NPW;
    if (start >= n) return;
    long end = start + NPW;
    if (end > n) end = n;

    int col = lane << 2;
    f4 acc = {0.f, 0.f, 0.f, 0.f};
    int cur = seg[start];

    for (long i = start; i < end; ++i) {
        int s = seg[i];                       // uniform across the wave
        if (s != cur) {
            float* p = pool + (long)cur * DIM + col;
            atomicAdd(p + 0, acc.x);
            atomicAdd(p + 1, acc.y);
            atomicAdd(p + 2, acc.z);
            atomicAdd(p + 3, acc.w);
            f4 z = {0.f, 0.f, 0.f, 0.f};
            acc = z;
            cur = s;
        }
        f4 hv = __builtin_nontemporal_load(
            (const f4*)(h + i * DIM + col));  // stream h, don't pollute L2
        acc += hv;
    }
    float* p = pool + (long)cur * DIM + col;
    atomicAdd(p + 0, acc.x);
    atomicAdd(p + 1, acc.y);
    atomicAdd(p + 2, acc.z);
    atomicAdd(p + 3, acc.w);
}

// ---------------- kernel 2: FC layer via V_WMMA_F32_16X16X4_F32 ------
// vn_out = vn_h + relu((vn_h + pool) @ W.T + b)
// One wave per 16x16 output tile; K-loop of 32 wmma (K=4 each).
__global__ void vn_gemm(const float* __restrict__ vn_h,
                        const float* __restrict__ pool,
                        const float* __restrict__ W,
                        const float* __restrict__ bias,
                        float* __restrict__ out_vn) {
    int wave = threadIdx.x >> 5;
    int lane = threadIdx.x & 31;
    int tile = blockIdx.x * 8 + wave;     // 1024 tiles total
    int mt = tile >> 3;                   // 128 M-tiles
    int nt = tile & 7;                    // 8  N-tiles
    int m0 = mt << 4;
    int n0 = nt << 4;
    int half = lane >> 4;                 // 0: K even pair low, 1: high
    int l16  = lane & 15;

    const float* xv = vn_h + (long)(m0 + l16) * DIM;  // A row (vn_h part)
    const float* xp = pool + (long)(m0 + l16) * DIM;  // A row (pool part)
    const float* wr = W    + (long)(n0 + l16) * DIM;  // B col = W row (y = x @ W.T)

    v8f acc = {};
    #pragma unroll
    for (int k0 = 0; k0 < DIM; k0 += 4) {
        int k = k0 + 2 * half;
        v2f a;
        a.x = xv[k]     + xp[k];          // X = vn_h + pool, on the fly
        a.y = xv[k + 1] + xp[k + 1];
        v2f bm = *(const v2f*)(wr + k);
        acc = __builtin_amdgcn_wmma_f32_16x16x4_f32(
            false, a, false, bm, (short)0, acc, false, false);
    }

    int n = n0 + l16;
    float bn = bias[n];
    #pragma unroll
    for (int r = 0; r < 8; ++r) {
        int m = m0 + half * 8 + r;        // C/D layout: VGPR r, lane half
        float y = acc[r] + bn;
        y = y > 0.f ? y : 0.f;            // relu
        long o = (long)m * DIM + n;
        out_vn[o] = vn_h[o] + y;          // residual on virtual node
    }
}

// ---------------- kernel 3: broadcast vn back to nodes ---------------
// h_out[i,:] = h[i,:] + vn_out[seg[i],:]   (vn_out is L2-resident, 1 MB)
__global__ void vn_scatter(const float* __restrict__ h,
                           const int* __restrict__ seg,
                           const float* __restrict__ vn_out,
                           float* __restrict__ h_out, int n) {
    long idx = (long)blockIdx.x * blockDim.x + threadIdx.x; // over n*32 float4s
    long row = idx >> 5;
    if (row >= n) return;
    int c = (int)(idx & 31) << 2;
    int s = seg[row];
    f4 hv = __builtin_nontemporal_load((const f4*)(h + row * DIM + c));
    f4 vv = *(const f4*)(vn_out + (long)s * DIM + c);   // cached gather
    f4 o = hv + vv;
    __builtin_nontemporal_store(o, (f4*)(h_out + row * DIM + c));
}

extern "C" void kernel_launch(void* const* d_in, const int* in_sizes, int n_in,
                              void* d_out, int out_size, void* d_ws, size_t ws_size,
                              hipStream_t stream) {
    const float* h    = (const float*)d_in[0];
    const float* vn_h = (const float*)d_in[1];
    const float* W    = (const float*)d_in[2];
    const float* b    = (const float*)d_in[3];
    const int*   seg  = (const int*)d_in[4];

    int n = in_sizes[0] / DIM;   // N nodes
    int B = in_sizes[1] / DIM;   // num graphs

    float* pool   = (float*)d_ws;                    // B*DIM f32 scratch
    float* out_vn = (float*)d_out;                   // [B, DIM]
    float* out_h  = (float*)d_out + (long)B * DIM;   // [N, DIM]

    // 0) zero pool (must re-zero every call; ws is not re-poisoned)
    int pn = B * DIM;
    vn_zero_pool<<<(pn + 255) / 256, 256, 0, stream>>>(pool, pn);

    // 1) segment sum
    int waves   = (n + NPW - 1) / NPW;
    int blocks1 = (waves + 7) / 8;                   // 8 waves / 256-thread block
    vn_segsum<<<blocks1, 256, 0, stream>>>(h, seg, pool, n);

    // 2) FC + relu + vn residual (WMMA): (B/16)*(DIM/16) tiles / 8 waves
    int tiles   = (B / 16) * (DIM / 16);
    vn_gemm<<<tiles / 8, 256, 0, stream>>>(vn_h, pool, W, b, out_vn);

    // 3) broadcast + node residual
    long units  = (long)n * 32;
    vn_scatter<<<(int)((units + 255) / 256), 256, 0, stream>>>(h, seg, out_vn, out_h, n);
}